// CustomLayer_3710851744539
// MI455X (gfx1250) — compile-verified
//
#include <hip/hip_runtime.h>

// ---------------------------------------------------------------------------
// Memristive crossbar passive solve on gfx1250 (MI455X).
//
// Structure-exploiting ADI / block Gauss-Seidel:
//   word-line half:  A_i VA_i = E_i + G_i o VB_i    (A_i: 64x64 tridiagonal)
//   bit-line  half:  D_j VB_j = G_j o VA_j          (D_j: 64x64 tridiagonal)
// Tridiagonal inverses precomputed (Thomas, diagonally dominant -> stable),
// each half-sweep becomes 64 independent (64x64)x(64x256) f32 GEMMs executed
// with V_WMMA_F32_16X16X4_F32. Coupling ratio G/g_line ~ 2e-3 => 4 sweeps
// reach fp32 round-off.
// ---------------------------------------------------------------------------

typedef __attribute__((ext_vector_type(2))) float v2f;
typedef __attribute__((ext_vector_type(8))) float v8f;

#define MDIM   64      // crossbar rows (inputs)
#define NDIM   64      // crossbar cols (outputs)
#define BATCH  256
#define G_LINE 0.5f    // 1 / PARASITIC_R, also g_in and g_out
#define CMAP   1.0e-3f // 1 / R_LRS mapping coefficient
#define NSWEEP 4

// ---------------------------------------------------------------------------
// Precompute dense inverses of the 64 word-line and 64 bit-line tridiagonal
// operators. One block per line; thread t computes column t of the inverse
// (Thomas solve with RHS = e_t). Shared forward coefficients in LDS.
// blocks 0..63  -> A_i   (word line i, diag G[i][j] + g*((j>0)+(j<63)) + g*(j==0))
// blocks 64..127-> D_j   (bit  line j, diag G[i][j] + g*((i>0)+(i<63)) + g*(i==63))
// off-diagonals are -g in both cases.  G[i][j] = CMAP * w[j*64 + i].
// ---------------------------------------------------------------------------
__global__ void __launch_bounds__(64)
build_inverses(const float* __restrict__ w, float* __restrict__ Ainv,
               float* __restrict__ Dinv)
{
    const float g = G_LINE;
    const int blk  = blockIdx.x;      // 0..127
    const int t    = threadIdx.x;     // 0..63 : inverse column
    const bool isA = (blk < 64);
    const int line = isA ? blk : blk - 64;

    __shared__ float cp[64];   // Thomas superdiagonal coefficients (shared)
    __shared__ float den[64];  // 1/denominator per step (shared)

    if (t == 0) {
        float cprev = 0.0f;
        for (int r = 0; r < 64; ++r) {
            // conductance of cell on this line at position r
            const float G = isA ? CMAP * w[r * MDIM + line]     // G[line][r]
                                : CMAP * w[line * MDIM + r];    // G[r][line]
            float d = G + g * ((r > 0 ? 1.0f : 0.0f) + (r < 63 ? 1.0f : 0.0f));
            if (isA)  d += (r == 0)  ? g : 0.0f;   // g_in at word-line entry
            else      d += (r == 63) ? g : 0.0f;   // g_out at bit-line exit
            const float denom = d + g * cprev;     // d - a*cp, a = -g
            const float inv = 1.0f / denom;
            den[r] = inv;
            cprev = (r < 63) ? (-g * inv) : 0.0f;  // cp_r = b_r/denom, b = -g
            cp[r] = cprev;
        }
    }
    __syncthreads();

    // forward sweep for RHS e_t (per-thread)
    float dp[64];
    float dprev = 0.0f;
#pragma unroll
    for (int r = 0; r < 64; ++r) {
        const float rhs = (r == t) ? 1.0f : 0.0f;
        dprev = (rhs + g * dprev) * den[r];        // (rhs - a*dp_{r-1})/denom
        dp[r] = dprev;
    }

    // back substitution; store column t of the inverse
    float* out = (isA ? Ainv : Dinv) + line * 64 * 64;
    float xv = 0.0f;
#pragma unroll
    for (int r = 63; r >= 0; --r) {
        xv = dp[r] - cp[r] * xv;
        out[r * 64 + t] = xv;                      // inv[row r][col t]
    }
}

__global__ void zero_kernel(float* __restrict__ p, int n)
{
    const int idx = blockIdx.x * blockDim.x + threadIdx.x;
    if (idx < n) p[idx] = 0.0f;
}

// ---------------------------------------------------------------------------
// Word-line half sweep:  VA[i] = Ainv_i @ ( g_in*x*delta(j==0) + G[i]oVB[i] )
// One wave per 16x16 output tile; K handled by 16 chained 16x16x4 f32 WMMAs.
// V layout: V[(i*64 + j)*BATCH + b].
// ---------------------------------------------------------------------------
__global__ void __launch_bounds__(128)
solve_word(const float* __restrict__ w, const float* __restrict__ x,
           const float* __restrict__ Ainv, const float* __restrict__ VB,
           float* __restrict__ VA)
{
    const int wave = blockIdx.x * (blockDim.x >> 5) + (threadIdx.x >> 5);
    const int lane = threadIdx.x & 31;
    const int bt = wave & 15;          // batch tile (16 cols)
    const int jt = (wave >> 4) & 3;    // row tile within line (16 rows)
    const int i  = wave >> 6;          // crossbar word line

    const int mrow = lane & 15;
    const int hi   = lane >> 4;        // 0: lanes 0-15, 1: lanes 16-31
    const int ksel = hi << 1;          // K offset 0 or 2 inside each 4-chunk

    const int bcol = bt * 16 + mrow;
    const float xin = G_LINE * x[bcol * MDIM + i];   // g_in * x[b][i]

    const float* Ai = Ainv + i * 64 * 64;
    v8f acc = {};
#pragma unroll
    for (int kk = 0; kk < 64; kk += 4) {
        const int ra = jt * 16 + mrow;               // output row (node j)
        v2f a;
        a.x = Ai[ra * 64 + kk + ksel + 0];
        a.y = Ai[ra * 64 + kk + ksel + 1];

        const int j0 = kk + ksel;                    // RHS node index
        const float vb0 = VB[(i * 64 + j0 + 0) * BATCH + bcol];
        const float vb1 = VB[(i * 64 + j0 + 1) * BATCH + bcol];
        v2f b;
        b.x = CMAP * w[(j0 + 0) * MDIM + i] * vb0 + ((j0 + 0) == 0 ? xin : 0.0f);
        b.y = CMAP * w[(j0 + 1) * MDIM + i] * vb1 + ((j0 + 1) == 0 ? xin : 0.0f);

        acc = __builtin_amdgcn_wmma_f32_16x16x4_f32(
            false, a, false, b, (short)0, acc, false, false);
    }

#pragma unroll
    for (int r = 0; r < 8; ++r) {
        const int row = jt * 16 + r + (hi << 3);
        VA[(i * 64 + row) * BATCH + bcol] = acc[r];
    }
}

// ---------------------------------------------------------------------------
// Bit-line half sweep:  VB[:,j] = Dinv_j @ ( G[:,j] o VA[:,j] )
// ---------------------------------------------------------------------------
__global__ void __launch_bounds__(128)
solve_bit(const float* __restrict__ w, const float* __restrict__ Dinv,
          const float* __restrict__ VA, float* __restrict__ VB)
{
    const int wave = blockIdx.x * (blockDim.x >> 5) + (threadIdx.x >> 5);
    const int lane = threadIdx.x & 31;
    const int bt = wave & 15;
    const int it = (wave >> 4) & 3;    // row tile (crossbar row index i)
    const int j  = wave >> 6;          // bit line

    const int mrow = lane & 15;
    const int hi   = lane >> 4;
    const int ksel = hi << 1;

    const int bcol = bt * 16 + mrow;

    const float* Dj = Dinv + j * 64 * 64;
    v8f acc = {};
#pragma unroll
    for (int kk = 0; kk < 64; kk += 4) {
        const int ra = it * 16 + mrow;
        v2f a;
        a.x = Dj[ra * 64 + kk + ksel + 0];
        a.y = Dj[ra * 64 + kk + ksel + 1];

        const int k0 = kk + ksel;                    // RHS crossbar-row index
        v2f b;
        b.x = CMAP * w[j * MDIM + k0 + 0] * VA[((k0 + 0) * 64 + j) * BATCH + bcol];
        b.y = CMAP * w[j * MDIM + k0 + 1] * VA[((k0 + 1) * 64 + j) * BATCH + bcol];

        acc = __builtin_amdgcn_wmma_f32_16x16x4_f32(
            false, a, false, b, (short)0, acc, false, false);
    }

#pragma unroll
    for (int r = 0; r < 8; ++r) {
        const int row = it * 16 + r + (hi << 3);
        VB[(row * 64 + j) * BATCH + bcol] = acc[r];
    }
}

// ---------------------------------------------------------------------------
// out[b][j] = bias[j] + sum_i w[j][i] * (VA[i][j][b] - VB[i][j][b])
// (division by CMAP folded: G/c = w)
// ---------------------------------------------------------------------------
__global__ void __launch_bounds__(256)
output_kernel(const float* __restrict__ w, const float* __restrict__ bias,
              const float* __restrict__ VA, const float* __restrict__ VB,
              float* __restrict__ out)
{
    const int j = blockIdx.x;      // output column
    const int b = threadIdx.x;     // batch
    float acc = bias[j];
#pragma unroll 8
    for (int i = 0; i < MDIM; ++i) {
        const int idx = (i * 64 + j) * BATCH + b;
        acc += w[j * MDIM + i] * (VA[idx] - VB[idx]);
    }
    out[b * NDIM + j] = acc;
}

extern "C" void kernel_launch(void* const* d_in, const int* in_sizes, int n_in,
                              void* d_out, int out_size, void* d_ws, size_t ws_size,
                              hipStream_t stream)
{
    (void)in_sizes; (void)n_in; (void)out_size; (void)ws_size;
    const float* x    = (const float*)d_in[0];   // (BATCH, 64)
    const float* w    = (const float*)d_in[1];   // (64, 64) weight[j][i]
    const float* bias = (const float*)d_in[2];   // (64)

    float* ws   = (float*)d_ws;
    float* Ainv = ws;                            // 64*64*64 f32  (1 MB)
    float* Dinv = Ainv + 64 * 64 * 64;           // 64*64*64 f32  (1 MB)
    float* VA   = Dinv + 64 * 64 * 64;           // 64*64*256 f32 (4 MB)
    float* VB   = VA + 64 * 64 * BATCH;          // 64*64*256 f32 (4 MB)
    float* out  = (float*)d_out;

    const int nV = 64 * 64 * BATCH;

    build_inverses<<<128, 64, 0, stream>>>(w, Ainv, Dinv);
    zero_kernel<<<(nV + 255) / 256, 256, 0, stream>>>(VB, nV);

    // 4096 tiles per half-sweep, 4 waves per block -> 1024 blocks
    for (int s = 0; s < NSWEEP; ++s) {
        solve_word<<<1024, 128, 0, stream>>>(w, x, Ainv, VB, VA);
        solve_bit <<<1024, 128, 0, stream>>>(w, Dinv, VA, VB);
    }
    // one extra word-line solve so VA is consistent with the final VB
    solve_word<<<1024, 128, 0, stream>>>(w, x, Ainv, VB, VA);

    output_kernel<<<64, 256, 0, stream>>>(w, bias, VA, VB, out);
}